// MultiDirectionPattern_55645596287450
// MI455X (gfx1250) — compile-verified
//
#include <hip/hip_runtime.h>
#include <hip/hip_bf16.h>

typedef __attribute__((ext_vector_type(2))) float v2f;
typedef __attribute__((ext_vector_type(8))) float v8f;

#define HW   49      // 7*7
#define KPAD 52      // pad K to multiple of 4 for 16x16x4 f32 WMMA
#define NP   8       // number of direction masks (output N)

// ---------------------------------------------------------------------------
// feat = (x[b,c,:,:] . mask[p,:,:]) / count[p], via V_WMMA_F32_16X16X4_F32.
// One wave computes a 16-row x 8-col output tile: 13 chained K=4 WMMAs.
// Memory-bound (103 MB of x read once, fully consumed per cache line).
// ---------------------------------------------------------------------------
__global__ __launch_bounds__(256) void feat_wmma_kernel(
    const float* __restrict__ x,     // (rows, 49) row-major, rows = B*C
    const float* __restrict__ fm,    // (8, 49) direction masks
    float* __restrict__ out,         // (rows, 8)
    int ntiles)                      // rows / 16
{
    __shared__ float bmat[KPAD * NP];   // B-matrix: bmat[k*8 + p], zero-padded K
    __shared__ float scount[NP];        // per-direction nonzero counts

    const int t = threadIdx.x;

    if (t < NP) {
        float c = 0.0f;
        for (int k = 0; k < HW; ++k)
            c += (fm[t * HW + k] != 0.0f) ? 1.0f : 0.0f;
        scount[t] = c;
    }
    for (int idx = t; idx < KPAD * NP; idx += 256) {
        int k = idx / NP, p = idx % NP;
        bmat[idx] = (k < HW) ? fm[p * HW + k] : 0.0f;
    }
    __syncthreads();

    const int lane = t & 31;
    const int wave = t >> 5;
    const int tile = blockIdx.x * 8 + wave;
    const bool valid = (tile < ntiles);
    const int tt = valid ? tile : (ntiles - 1);   // clamp: keep loads in-bounds,
                                                  // EXEC all-ones through WMMAs
    const long row_base = (long)tt * 16;

    const int m  = lane & 15;          // A: row within tile; B/D: column N
    const int hi = (lane >> 4) * 2;    // K sub-offset for upper lane half

    const float* xrow = x + (row_base + m) * (long)HW;

    v8f acc = {};

    // K = 0..47 : 12 full 16x16x4 f32 WMMAs
    for (int kc = 0; kc < 12; ++kc) {
        const int k0 = kc * 4 + hi;
        v2f a, b;
        a.x = xrow[k0];
        a.y = xrow[k0 + 1];
        float b0 = bmat[k0 * NP + (m & 7)];
        float b1 = bmat[(k0 + 1) * NP + (m & 7)];
        b.x = (m < NP) ? b0 : 0.0f;    // B columns 8..15 are zero
        b.y = (m < NP) ? b1 : 0.0f;
        acc = __builtin_amdgcn_wmma_f32_16x16x4_f32(
            false, a, false, b, (short)0, acc, false, false);
    }

    // K tail = 48..51 : only K=48 is real, rest zero
    {
        v2f a, b;
        a.x = (hi == 0) ? xrow[48] : 0.0f;
        a.y = 0.0f;
        b.x = (hi == 0 && m < NP) ? bmat[48 * NP + m] : 0.0f;
        b.y = 0.0f;
        acc = __builtin_amdgcn_wmma_f32_16x16x4_f32(
            false, a, false, b, (short)0, acc, false, false);
    }

    // D layout: VGPR v -> (M = v + (lane>=16 ? 8 : 0), N = lane&15)
    if (valid && m < NP) {
        const float cnt = scount[m];
        const int rlo = (lane >> 4) * 8;
        #pragma unroll
        for (int v = 0; v < 8; ++v) {
            const long row = row_base + v + rlo;
            out[row * NP + m] = acc[v] / cnt;
        }
    }
}

// ---------------------------------------------------------------------------
// sal_feat[b, n, p] = s[b, sal_idx[p, n]]  (output-coalesced gather; s row is
// 12.5 KB and sal_idx is ~13 KB -> both L2/L0 resident)
// ---------------------------------------------------------------------------
__global__ __launch_bounds__(256) void sal_gather_kernel(
    const float* __restrict__ s,        // (B, 3136)
    const int*   __restrict__ sal_idx,  // (8, N)
    float* __restrict__ out,            // (B, N, 8)
    int B, int N, int spb)              // spb = 56*56
{
    const long total = (long)B * N * NP;
    const long tid = (long)blockIdx.x * 256 + threadIdx.x;
    if (tid >= total) return;
    const int p = (int)(tid % NP);
    const long r = tid / NP;
    const int n = (int)(r % N);
    const int b = (int)(r / N);
    out[tid] = s[(long)b * spb + sal_idx[p * N + n]];
}

extern "C" void kernel_launch(void* const* d_in, const int* in_sizes, int n_in,
                              void* d_out, int out_size, void* d_ws, size_t ws_size,
                              hipStream_t stream) {
    const float* x       = (const float*)d_in[0];   // (B, C, 7, 7)
    const float* s       = (const float*)d_in[1];   // (B, 1, 56, 56)
    const float* fm      = (const float*)d_in[2];   // (8, 7, 7)
    const int*   sal_idx = (const int*)d_in[3];     // (8, N)

    const int spb = 56 * 56;
    const int B = in_sizes[1] / spb;                // 256
    const int C = in_sizes[0] / (B * HW);           // 2048
    const int N = in_sizes[3] / NP;

    float* out_feat = (float*)d_out;                // (B*C, 8)
    const long rows = (long)B * C;                  // 524288 -> /16 exact
    float* out_sal = out_feat + rows * NP;          // (B, N, 8)

    const int ntiles = (int)(rows / 16);
    const int feat_blocks = (ntiles + 7) / 8;       // 8 waves (wave32) per block
    feat_wmma_kernel<<<feat_blocks, 256, 0, stream>>>(x, fm, out_feat, ntiles);

    const long total = (long)B * N * NP;
    const int gather_blocks = (int)((total + 255) / 256);
    sal_gather_kernel<<<gather_blocks, 256, 0, stream>>>(s, sal_idx, out_sal, B, N, spb);
}